// GraphAttentionLayer_9655086481445
// MI455X (gfx1250) — compile-verified
//
#include <hip/hip_runtime.h>

typedef __attribute__((ext_vector_type(2))) float v2f;
typedef __attribute__((ext_vector_type(8))) float v8f;

#define B_ 2
#define N_ 512
#define D_ 256
#define H_ 4
#define HD_ 64
#define SLOPE 0.2f

// ---------------------------------------------------------------------------
// Kernel 1: projections hq = q@Wq+bq, hk = k@Wq+bq, hv = v@Wv+bv
// via V_WMMA_F32_16X16X4_F32. One wave per 16x16 output tile.
// 6 matrices (B * {q,k,v}) * (32 row tiles * 16 col tiles) = 3072 waves.
// ---------------------------------------------------------------------------
__global__ __launch_bounds__(256) void gat_proj_wmma(
    const float* __restrict__ query, const float* __restrict__ key_in,
    const float* __restrict__ value, const float* __restrict__ w_q,
    const float* __restrict__ b_q,   const float* __restrict__ w_v,
    const float* __restrict__ b_v,   float* __restrict__ h_ws) {
  const int lane = threadIdx.x & 31;
  const int wid  = (blockIdx.x * blockDim.x + threadIdx.x) >> 5;

  const int mat = wid >> 9;        // 0..5  (b*3 + sel)
  const int t   = wid & 511;
  const int b   = mat / 3;
  const int sel = mat % 3;         // 0=q, 1=k, 2=v
  const int tm  = t >> 4;          // 0..31 row tile
  const int tn  = t & 15;          // 0..15 col tile

  const float* X    = (sel == 0) ? query : (sel == 1) ? key_in : value;
  const float* W    = (sel == 2) ? w_v : w_q;
  const float* bias = (sel == 2) ? b_v : b_q;
  float* out = h_ws + (size_t)sel * (B_ * N_ * D_) + (size_t)b * (N_ * D_);

  const int half = lane >> 4;      // 0 or 1 (lane group)
  const int l16  = lane & 15;
  const int row  = tm * 16 + l16;  // A-matrix row this lane supplies
  const int col  = tn * 16 + l16;  // B-matrix column this lane supplies

  const float* Xrow = X + (size_t)b * (N_ * D_) + (size_t)row * D_;

  v8f acc = {};
  for (int k0 = 0; k0 < D_; k0 += 4) {
    const int ka = k0 + 2 * half;               // this lane's K pair: ka, ka+1
    // A 16x4 layout: lane holds M=lane%16, K = 2*(lane/16) + {0,1} -> contiguous
    v2f av = *(const v2f*)(Xrow + ka);
    // B 4x16 layout (symmetric): lane holds N=lane%16, K = 2*(lane/16) + {0,1}
    v2f bv;
    bv.x = W[(size_t)(ka + 0) * D_ + col];
    bv.y = W[(size_t)(ka + 1) * D_ + col];
    acc = __builtin_amdgcn_wmma_f32_16x16x4_f32(
        /*neg_a=*/false, av, /*neg_b=*/false, bv,
        /*c_mod=*/(short)0, acc, /*reuse_a=*/false, /*reuse_b=*/false);
  }

  const float bb = bias[col];
#pragma unroll
  for (int r = 0; r < 8; ++r) {
    // C/D layout: VGPR r -> row r (lanes 0-15) or r+8 (lanes 16-31), col = lane%16
    const int m = tm * 16 + r + 8 * half;
    out[(size_t)m * D_ + col] = acc[r] + bb;
  }
}

// ---------------------------------------------------------------------------
// Kernel 2: raw attention scores (LeakyReLU inside contraction -> VALU).
// Block = (b, h, 16-row i-tile); 256 threads, thread (ii,jj) owns one (i,j).
// Writes scores straight into e output region, reference layout [b,i,j,h].
// ---------------------------------------------------------------------------
__global__ __launch_bounds__(256) void gat_scores(
    const float* __restrict__ h_ws, const float* __restrict__ a_vec,
    float* __restrict__ e_out) {
  const int blk = blockIdx.x;            // B*H*32 = 256 blocks
  const int b   = blk >> 7;
  const int h   = (blk >> 5) & 3;
  const int it  = blk & 31;
  const int tid = threadIdx.x;

  const float* hq = h_ws + (size_t)b * (N_ * D_);
  const float* hk = h_ws + (size_t)(B_ * N_ * D_) + (size_t)b * (N_ * D_);

  __shared__ float hks[16 * HD_];
  __shared__ float hqs[16 * HD_];
  __shared__ float as[HD_];

  if (tid < HD_) as[tid] = a_vec[tid];
  for (int x = tid; x < 16 * HD_; x += 256) {
    const int r = x >> 6, f = x & 63;
    hks[x] = hk[(size_t)(it * 16 + r) * D_ + h * HD_ + f];
  }

  const int ii = tid >> 4, jj = tid & 15;
  const int i  = it * 16 + ii;

  for (int j0 = 0; j0 < N_; j0 += 16) {
    __syncthreads();                     // also covers hks/as first use
    for (int x = tid; x < 16 * HD_; x += 256) {
      const int r = x >> 6, f = x & 63;
      hqs[x] = hq[(size_t)(j0 + r) * D_ + h * HD_ + f];
    }
    __syncthreads();
    float s = 0.f;
#pragma unroll 8
    for (int f = 0; f < HD_; ++f) {
      const float x  = hqs[jj * HD_ + f] + hks[ii * HD_ + f];
      const float lr = fmaxf(x, 0.f) + SLOPE * fminf(x, 0.f);
      s += lr * as[f];
    }
    e_out[(((size_t)b * N_ + i) * N_ + (j0 + jj)) * H_ + h] = s;
  }
}

// ---------------------------------------------------------------------------
// Kernel 3: softmax over j for each (b,i,h) row, in place (stride H).
// One wave per row; 4096 rows; shfl-xor reductions (wave32).
// ---------------------------------------------------------------------------
__global__ __launch_bounds__(128) void gat_softmax(float* __restrict__ e_out) {
  const int wid  = (blockIdx.x * blockDim.x + threadIdx.x) >> 5;  // 0..4095
  const int lane = threadIdx.x & 31;
  const int b    = wid >> 11;
  const int rem  = wid & 2047;
  const int i    = rem >> 2;
  const int h    = rem & 3;

  float* row = e_out + (((size_t)b * N_ + i) * N_) * H_ + h;

  float vals[16];
  float m = -3.402823466e38f;
#pragma unroll
  for (int t = 0; t < 16; ++t) {
    const float v = row[(size_t)(lane + t * 32) * H_];
    vals[t] = v;
    m = fmaxf(m, v);
  }
#pragma unroll
  for (int o = 16; o > 0; o >>= 1) m = fmaxf(m, __shfl_xor(m, o, 32));

  float sum = 0.f;
#pragma unroll
  for (int t = 0; t < 16; ++t) {
    vals[t] = __expf(vals[t] - m);
    sum += vals[t];
  }
#pragma unroll
  for (int o = 16; o > 0; o >>= 1) sum += __shfl_xor(sum, o, 32);

  const float inv = 1.0f / sum;
#pragma unroll
  for (int t = 0; t < 16; ++t) row[(size_t)(lane + t * 32) * H_] = vals[t] * inv;
}

// ---------------------------------------------------------------------------
// Kernel 4: out[b,i,h,:] = relu( e[b,i,:,h] @ hv[b,:,h,:] ) via fp32 WMMA.
// Per (b,h): 512x512 @ 512x64. One wave per 16x16 output tile -> 1024 waves.
// ---------------------------------------------------------------------------
__global__ __launch_bounds__(256) void gat_aggregate_wmma(
    const float* __restrict__ e_out, const float* __restrict__ h_ws,
    float* __restrict__ out_hp) {
  const int lane = threadIdx.x & 31;
  const int wid  = (blockIdx.x * blockDim.x + threadIdx.x) >> 5;  // 0..1023

  const int bh = wid >> 7;         // 0..7
  const int t  = wid & 127;
  const int b  = bh >> 2, h = bh & 3;
  const int tm = t >> 2;           // 0..31 (i tile)
  const int tn = t & 3;            // 0..3  (f tile)

  const int half = lane >> 4;
  const int l16  = lane & 15;
  const int i    = tm * 16 + l16;  // A row
  const int f    = tn * 16 + l16;  // B column (within head)

  const float* hv = h_ws + (size_t)2 * (B_ * N_ * D_) + (size_t)b * (N_ * D_);
  const float* P  = e_out + (((size_t)b * N_ + i) * N_) * H_ + h;  // stride H_

  v8f acc = {};
  for (int k0 = 0; k0 < N_; k0 += 4) {
    const int k = k0 + 2 * half;
    v2f av;
    av.x = P[(size_t)(k + 0) * H_];
    av.y = P[(size_t)(k + 1) * H_];
    v2f bv;
    bv.x = hv[(size_t)(k + 0) * D_ + h * HD_ + f];
    bv.y = hv[(size_t)(k + 1) * D_ + h * HD_ + f];
    acc = __builtin_amdgcn_wmma_f32_16x16x4_f32(
        false, av, false, bv, (short)0, acc, false, false);
  }

#pragma unroll
  for (int r = 0; r < 8; ++r) {
    const int m = tm * 16 + r + 8 * half;
    out_hp[((size_t)b * N_ + m) * D_ + h * HD_ + f] = fmaxf(acc[r], 0.f);
  }
}

// ---------------------------------------------------------------------------
extern "C" void kernel_launch(void* const* d_in, const int* in_sizes, int n_in,
                              void* d_out, int out_size, void* d_ws, size_t ws_size,
                              hipStream_t stream) {
  const float* query  = (const float*)d_in[0];
  const float* key_in = (const float*)d_in[1];
  const float* value  = (const float*)d_in[2];
  const float* w_q    = (const float*)d_in[3];
  const float* b_q    = (const float*)d_in[4];
  const float* w_v    = (const float*)d_in[5];
  const float* b_v    = (const float*)d_in[6];
  const float* a_vec  = (const float*)d_in[7];

  float* out_hp = (float*)d_out;                       // [B, N, H*HD]
  float* out_e  = out_hp + (size_t)B_ * N_ * H_ * HD_; // [B, N, N, H]
  float* h_ws   = (float*)d_ws;                        // hq | hk | hv (3 MB)

  // 1) projections: 3072 waves = 384 blocks * 8 waves
  gat_proj_wmma<<<384, 256, 0, stream>>>(query, key_in, value,
                                         w_q, b_q, w_v, b_v, h_ws);
  // 2) raw scores -> e region (layout [b,i,j,h])
  gat_scores<<<B_ * H_ * (N_ / 16), 256, 0, stream>>>(h_ws, a_vec, out_e);
  // 3) softmax over j, in place: 4096 rows, 4 waves per block
  gat_softmax<<<1024, 128, 0, stream>>>(out_e);
  // 4) aggregation GEMM + ReLU: 1024 waves = 128 blocks * 8 waves
  gat_aggregate_wmma<<<128, 256, 0, stream>>>(out_e, h_ws, out_hp);
}